// WaveletTransform_14319420965150
// MI455X (gfx1250) — compile-verified
//
#include <hip/hip_runtime.h>
#include <stdint.h>

// ---------------------------------------------------------------------------
// 2D Haar DWT, [8,32,512,512] f32 -> [8,4,32,256,256] f32.
// Roofline: 256MiB in + 256MiB out, 0.27 GFLOP => 0.5 flop/byte, HBM-bound;
// floor ~23us at 23.3 TB/s. Strategy: stream at full rate.
//   - GLOBAL_LOAD_ASYNC_TO_LDS_B128 (ASYNCcnt) DMA staging of 8x512 tiles
//   - s_wait_asynccnt + barrier, ds_load_b128 tile reads
//   - butterfly VALU (8 add + 4 mul per output quad)
//   - non-temporal B128 stores (write-once) and NT async loads (read-once):
//     512MiB of streamed data must not thrash the 192MB L2.
// WMMA deliberately unused: a 4x4 Haar butterfly via V_WMMA_F32_16X16X4_F32
// would add lane-layout shuffles costing more than the FLOPs it replaces,
// with zero effect on the HBM-bound runtime.
// ---------------------------------------------------------------------------

typedef float v4f __attribute__((ext_vector_type(4)));
typedef int   v4i __attribute__((vector_size(16)));   // matches builtin param

#ifndef __has_builtin
#define __has_builtin(x) 0
#endif

#define CPOL_TH_NT 1   // gfx12 CPol: TH bits [2:0]; 1 = non-temporal

// Async DMA of 16 bytes: global -> LDS. Tracked by ASYNCcnt.
__device__ __forceinline__ void async_copy_b128(const float* gp, uint32_t lds_byte_addr) {
#if __has_builtin(__builtin_amdgcn_global_load_async_to_lds_b128)
    typedef __attribute__((address_space(1))) v4i* g_v4i_p;   // global int4*
    typedef __attribute__((address_space(3))) v4i* l_v4i_p;   // LDS int4*
    __builtin_amdgcn_global_load_async_to_lds_b128(
        (g_v4i_p)(uintptr_t)gp,
        (l_v4i_p)lds_byte_addr,            // AS3 ptr == 32-bit LDS byte offset
        /*imm offset*/0, /*imm cpol*/CPOL_TH_NT);
#else
    // GV addressing: vdst = LDS byte address, vaddr = 64-bit global address.
    asm volatile("global_load_async_to_lds_b128 %0, %1, off th:TH_LOAD_NT"
                 :: "v"(lds_byte_addr), "v"((uint64_t)(uintptr_t)gp)
                 : "memory");
#endif
}

__device__ __forceinline__ void wait_async_all() {
#if __has_builtin(__builtin_amdgcn_s_wait_asynccnt)
    __builtin_amdgcn_s_wait_asynccnt(0);
#else
    asm volatile("s_wait_asynccnt 0x0" ::: "memory");
#endif
}

#define W_IN     512
#define PLANE_IN (512 * 512)
#define W2       256
#define PLANE_O  (256 * 256)            // per (b,k,c) plane
#define K_STRIDE ((size_t)32 * PLANE_O) // stride between subband planes
#define LDS_ROW_F 516                   // 512 floats + 16B pad (keeps 16B align)

__global__ __launch_bounds__(256)
void haar_dwt2_kernel(const float* __restrict__ in, float* __restrict__ out) {
    // 8 input rows x 512 floats staged per block (16.5 KB of 320 KB LDS/WGP).
    __shared__ __attribute__((aligned(16))) float tile[8 * LDS_ROW_F];

    const int t  = threadIdx.x;     // 0..255
    const int hg = blockIdx.x;      // 0..63  : group of 4 output rows
    const int bc = blockIdx.y;      // 0..255 : b*32 + c

    // ---- Stage 8x512 input tile into LDS via async DMA (4x B128 per thread)
    const float*   g0   = in + (size_t)bc * PLANE_IN + (size_t)hg * (8 * W_IN);
    const uint32_t lds0 = (uint32_t)(uintptr_t)tile;
#pragma unroll
    for (int j = 0; j < 4; ++j) {
        const int idx = t + 256 * j;        // 0..1023 float4 slots (8 rows x 128)
        const int row = idx >> 7;
        const int col = idx & 127;
        async_copy_b128(g0 + (size_t)idx * 4,
                        lds0 + (uint32_t)(row * (LDS_ROW_F * 4) + col * 16));
    }
    wait_async_all();      // this wave's async copies landed in LDS
    __syncthreads();       // all waves' copies visible

    // ---- Compute: thread owns output row r, 4 output columns starting w4*4
    const int r  = t >> 6;                  // 0..3
    const int w4 = t & 63;                  // 0..63
    const float* l0 = &tile[(2 * r) * LDS_ROW_F + w4 * 8];
    const float* l1 = l0 + LDS_ROW_F;       // odd input row

    const v4f A0 = *(const v4f*)(l0);       // row0: cols 8w4..8w4+3
    const v4f A1 = *(const v4f*)(l0 + 4);   // row0: cols 8w4+4..8w4+7
    const v4f B0 = *(const v4f*)(l1);
    const v4f B1 = *(const v4f*)(l1 + 4);

    v4f ll, lh, hl, hh;
    {   // butterfly: s=even+odd, d=even-odd per row, then combine rows
        float s0, d0, s1, d1;
        s0 = A0.x + A0.y; d0 = A0.x - A0.y; s1 = B0.x + B0.y; d1 = B0.x - B0.y;
        ll.x = (s0 + s1) * 0.5f; lh.x = (s0 - s1) * 0.5f;
        hl.x = (d0 + d1) * 0.5f; hh.x = (d0 - d1) * 0.5f;

        s0 = A0.z + A0.w; d0 = A0.z - A0.w; s1 = B0.z + B0.w; d1 = B0.z - B0.w;
        ll.y = (s0 + s1) * 0.5f; lh.y = (s0 - s1) * 0.5f;
        hl.y = (d0 + d1) * 0.5f; hh.y = (d0 - d1) * 0.5f;

        s0 = A1.x + A1.y; d0 = A1.x - A1.y; s1 = B1.x + B1.y; d1 = B1.x - B1.y;
        ll.z = (s0 + s1) * 0.5f; lh.z = (s0 - s1) * 0.5f;
        hl.z = (d0 + d1) * 0.5f; hh.z = (d0 - d1) * 0.5f;

        s0 = A1.z + A1.w; d0 = A1.z - A1.w; s1 = B1.z + B1.w; d1 = B1.z - B1.w;
        ll.w = (s0 + s1) * 0.5f; lh.w = (s0 - s1) * 0.5f;
        hl.w = (d0 + d1) * 0.5f; hh.w = (d0 - d1) * 0.5f;
    }

    // ---- Non-temporal B128 stores, one per subband plane (write-once data)
    const int b  = bc >> 5;
    const int c  = bc & 31;
    const int oh = hg * 4 + r;              // output row 0..255
    float* op = out + ((size_t)(b * 4) * 32 + c) * PLANE_O
                    + (size_t)oh * W2 + w4 * 4;
    __builtin_nontemporal_store(ll, (v4f*)(op));
    __builtin_nontemporal_store(lh, (v4f*)(op + K_STRIDE));
    __builtin_nontemporal_store(hl, (v4f*)(op + 2 * K_STRIDE));
    __builtin_nontemporal_store(hh, (v4f*)(op + 3 * K_STRIDE));
}

extern "C" void kernel_launch(void* const* d_in, const int* in_sizes, int n_in,
                              void* d_out, int out_size, void* d_ws, size_t ws_size,
                              hipStream_t stream) {
    (void)in_sizes; (void)n_in; (void)out_size; (void)d_ws; (void)ws_size;
    const float* in  = (const float*)d_in[0];
    float*       out = (float*)d_out;
    // grid.x = 64 row-groups, grid.y = 256 (b*32+c) planes; 256 thr = 8 waves
    dim3 grid(64, 256, 1);
    haar_dwt2_kernel<<<grid, dim3(256, 1, 1), 0, stream>>>(in, out);
}